// JJ_Norm_21474836480017
// MI455X (gfx1250) — compile-verified
//
#include <hip/hip_runtime.h>
#include <hip/hip_bf16.h>

// Problem constants (from setup_inputs)
#define T_ 20
#define L_ 40
#define SPLIT_ 15
#define G_ (T_*L_)       // 800 (time,label) groups
#define D_ 128
#define DSLICE 32        // dims per pass-1 slice block
#define NSLICE (D_/DSLICE)

// Pass-1 LDS layout (floats)
#define SM_ACC   (G_*DSLICE + 2*G_ + DSLICE + 2)   // accumulators: 27234 floats
#define STAGE_OFF (((SM_ACC) + 15) & ~15)          // 64B-align stage: 27248 floats
#define STAGE_FLOATS (8 * 2 * 128)                 // 8 waves x 2 bufs x 512B
#define SM_TOTAL (STAGE_OFF + STAGE_FLOATS)        // 29296 floats (~117 KB)

typedef __attribute__((ext_vector_type(2))) float v2f;
typedef __attribute__((ext_vector_type(4))) float v4f;
typedef __attribute__((ext_vector_type(8))) float v8f;

#if defined(__has_builtin)
#if __has_builtin(__builtin_amdgcn_wmma_f32_16x16x4_f32)
#define HAVE_WMMA_F32 1
#endif
#endif

__device__ __forceinline__ void gAdd(float* p, float v) {
  __hip_atomic_fetch_add(p, v, __ATOMIC_RELAXED, __HIP_MEMORY_SCOPE_AGENT);     // global_atomic_add_f32
}
__device__ __forceinline__ void sAdd(float* p, float v) {
  __hip_atomic_fetch_add(p, v, __ATOMIC_RELAXED, __HIP_MEMORY_SCOPE_WORKGROUP); // ds_add_f32
}

// Per-lane async copy: 16B of x for row-quad q into the wave's LDS stage buffer.
// ASYNCcnt-tracked; does not consume VGPR return slots or LOADcnt.
__device__ __forceinline__ void issue_async_quad(
    const float* __restrict__ x, int N, int q, int k, int dbase, int j,
    unsigned stage_base_bytes, int buf)
{
  int row = 4*q + k;
  if (row >= N) row = N - 1;                      // clamp (zeroed at consume)
  const float* g = x + (size_t)row * D_ + dbase + 4*j;
  const unsigned loff = stage_base_bytes + (unsigned)(buf * 512 + (threadIdx.x & 31) * 16);
  asm volatile("global_load_async_to_lds_b128 %0, %1, off"
               :: "v"(loff), "v"(g) : "memory");
}

// ---------------------------------------------------------------------------
// Pass 1: streaming moment accumulation over x.
// grid = (row-chunks, NSLICE). Each block owns a 32-dim slice and accumulates
// per-(t,l) vector sums, per-(t,l) sum of squared norms, counts, and the test
// vector sum / scalar moments in LDS; flushed once with global fp atomics.
// x is streamed through a per-wave async double buffer in LDS.
// ---------------------------------------------------------------------------
__global__ __launch_bounds__(256) void jj_norm_p1(
    const float* __restrict__ x, const int* __restrict__ labels,
    const int* __restrict__ times, int N,
    float* __restrict__ gsums, float* __restrict__ gsumsq,
    float* __restrict__ gcnt, float* __restrict__ gtsum,
    float* __restrict__ gtsc)
{
  extern __shared__ float smem[];
  float* s_sums  = smem;                    // G_*DSLICE
  float* s_sumsq = s_sums + G_*DSLICE;      // G_
  float* s_cnt   = s_sumsq + G_;            // G_  (slice 0 only)
  float* s_tsum  = s_cnt + G_;              // DSLICE
  float* s_tsc   = s_tsum + DSLICE;         // [0]=test_cnt (slice 0), [1]=test sum||x||^2 (partial)
  float* s_stage = smem + STAGE_OFF;        // 8 waves x 2 bufs x 128 floats
  const int tid = threadIdx.x;
  for (int i = tid; i < SM_ACC; i += blockDim.x) smem[i] = 0.0f;
  __syncthreads();

  const int slice = blockIdx.y;
  const int dbase = slice * DSLICE;
  const int lane  = tid & 31;
  const int wv    = tid >> 5;     // wave in block (8 waves)
  const int j     = lane & 7;     // float4 quad within slice
  const int k     = lane >> 3;    // row within row-quad
  const int numQ  = (N + 3) >> 2;
  const int qstride = gridDim.x * 8;

  float* stage = s_stage + wv * 256;                          // this wave's 2 buffers
  const unsigned stage_base = (unsigned)(uintptr_t)(void*)stage; // low 32 bits = LDS byte offset

  const int q0 = blockIdx.x * 8 + wv;
  if (q0 < numQ)
    issue_async_quad(x, N, q0, k, dbase, j, stage_base, 0);   // prologue

  int buf = 0;
  for (int q = q0; q < numQ; q += qstride, buf ^= 1) {
    const int qn = q + qstride;
    if (qn < numQ) {
      issue_async_quad(x, N, qn, k, dbase, j, stage_base, buf ^ 1);
      // warm L2 one more stride out
      if (4*(qn + qstride) + k < N)
        __builtin_prefetch(x + (size_t)(4*(qn + qstride) + k) * D_ + dbase + 4*j, 0, 0);
      asm volatile("s_wait_asynccnt 0x1" ::: "memory");       // current buffer ready
    } else {
      asm volatile("s_wait_asynccnt 0x0" ::: "memory");
    }

    const int row = 4*q + k;
    const bool valid = row < N;
    v4f xv = *(const v4f*)(stage + buf * 128 + lane * 4);     // ds_load_b128
    if (!valid) { xv[0] = 0.f; xv[1] = 0.f; xv[2] = 0.f; xv[3] = 0.f; }
    int t = 0, l = 0;
    if (valid) { t = times[row]; l = labels[row]; }

    float r = xv[0]*xv[0] + xv[1]*xv[1] + xv[2]*xv[2] + xv[3]*xv[3];
    r += __shfl_xor(r, 1, 32);
    r += __shfl_xor(r, 2, 32);
    r += __shfl_xor(r, 4, 32);   // sum of x^2 over this row's 32-dim slice

    if (valid) {
      if (t < SPLIT_) {
        const int seg = t * L_ + l;
        float* b = &s_sums[seg * DSLICE + 4*j];   // consecutive banks, conflict-free
        sAdd(b+0, xv[0]); sAdd(b+1, xv[1]); sAdd(b+2, xv[2]); sAdd(b+3, xv[3]);
        if (j == 0) {
          sAdd(&s_sumsq[seg], r);
          if (slice == 0) sAdd(&s_cnt[seg], 1.0f);
        }
      } else {
        float* b = &s_tsum[4*j];
        sAdd(b+0, xv[0]); sAdd(b+1, xv[1]); sAdd(b+2, xv[2]); sAdd(b+3, xv[3]);
        if (j == 0) {
          sAdd(&s_tsc[1], r);
          if (slice == 0) sAdd(&s_tsc[0], 1.0f);
        }
      }
    }
  }
  __syncthreads();
  for (int i = tid; i < G_*DSLICE; i += blockDim.x)
    gAdd(&gsums[(i / DSLICE) * D_ + dbase + (i % DSLICE)], s_sums[i]);
  for (int i = tid; i < G_; i += blockDim.x) {
    gAdd(&gsumsq[i], s_sumsq[i]);
    if (slice == 0) gAdd(&gcnt[i], s_cnt[i]);
  }
  for (int i = tid; i < DSLICE; i += blockDim.x)
    gAdd(&gtsum[dbase + i], s_tsum[i]);
  if (tid == 0) {
    gAdd(&gtsc[1], s_tsc[1]);
    if (slice == 0) gAdd(&gtsc[0], s_tsc[0]);
  }
}

// ---------------------------------------------------------------------------
// Pass 2 (single block): means, time-means, msq/rsq via closed forms, alpha.
// Per-group squared norms computed as diagonals of mu*mu^T tiles with
// V_WMMA_F32_16X16X4_F32 (A and B share the same lane data for A*A^T).
// ---------------------------------------------------------------------------
__global__ __launch_bounds__(256) void jj_norm_p2(
    const float* __restrict__ gsums, const float* __restrict__ gsumsq,
    const float* __restrict__ gcnt, const float* __restrict__ gtsum,
    const float* __restrict__ gtsc, float* __restrict__ gmean,
    float* __restrict__ galpha)
{
  __shared__ float s_tmean[T_*D_];
  __shared__ float s_tcnt[T_];
  __shared__ float s_msq[T_], s_rA[T_], s_rB[T_];
  __shared__ float s_red[256];
  const int tid = threadIdx.x;

  if (tid < T_) {
    s_msq[tid] = 0.f; s_rB[tid] = 0.f;
    float c = 0.f, a = 0.f;
    for (int l = 0; l < L_; ++l) { c += gcnt[tid*L_ + l]; a += gsumsq[tid*L_ + l]; }
    s_tcnt[tid] = c;      // train rows at time t
    s_rA[tid]   = a;      // sum over groups of sum ||x||^2
  }
  __syncthreads();

  for (int i = tid; i < T_*D_; i += blockDim.x) {       // time means
    const int t = i / D_, d = i - t*D_;
    float s = 0.f;
    for (int l = 0; l < L_; ++l) s += gsums[(t*L_ + l)*D_ + d];
    s_tmean[i] = s / fmaxf(1.f, s_tcnt[t]);
  }
  for (int i = tid; i < G_*D_; i += blockDim.x)          // group means -> global (pass 3)
    gmean[i] = gsums[i] / fmaxf(1.f, gcnt[i / D_]);
  __syncthreads();

#if HAVE_WMMA_F32
  {
    const int wv = tid >> 5, lane = tid & 31;
    const int gi0 = lane & 15;        // A row M / B col N within tile
    const int khalf = lane >> 4;      // lanes 16-31 carry K=2,3
    for (int tile = wv; tile < G_/16; tile += 8) {       // 50 tiles of 16 groups
      const int gi = tile*16 + gi0;
      const int t  = gi / L_;
      const float invc = 1.f / fmaxf(1.f, gcnt[gi]);
      v8f cm = {}; v8f cr = {};
      for (int kb = 0; kb < D_; kb += 4) {
        const int kk = kb + 2*khalf;
        const float m0 = gsums[gi*D_ + kk]     * invc;
        const float m1 = gsums[gi*D_ + kk + 1] * invc;
        v2f am; am[0] = m0; am[1] = m1;                          // mu
        v2f ad; ad[0] = m0 - s_tmean[t*D_ + kk];                  // mu - tau
        ad[1] = m1 - s_tmean[t*D_ + kk + 1];
        // D += A * A^T ; diagonal = partial squared norms
        cm = __builtin_amdgcn_wmma_f32_16x16x4_f32(false, ad, false, ad, (short)0, cm, false, false);
        cr = __builtin_amdgcn_wmma_f32_16x16x4_f32(false, am, false, am, (short)0, cr, false, false);
      }
      // C layout: VGPR r -> (M=r, N=lane) for lanes 0-15; (M=r+8, N=lane-16) for 16-31.
      int m = -1;
      if (lane < 8)        m = lane;        // (m,m) in cm[lane]
      else if (lane >= 24) m = lane - 16;   // (m,m) in cm[lane-24]
      if (m >= 0) {
        const int r  = m & 7;
        const int g  = tile*16 + m;
        const float c = gcnt[g];
        const int tg = g / L_;
        sAdd(&s_msq[tg], c * cm[r]);        // sum_l cnt*||mu - tau||^2
        sAdd(&s_rB[tg],  c * cr[r]);        // sum_l cnt*||mu||^2
      }
    }
  }
#else
  for (int g = tid; g < G_; g += blockDim.x) {
    const int t = g / L_;
    const float invc = 1.f / fmaxf(1.f, gcnt[g]);
    float nm = 0.f, nr = 0.f;
    for (int d = 0; d < D_; ++d) {
      const float m  = gsums[g*D_ + d] * invc;
      const float df = m - s_tmean[t*D_ + d];
      nm += df*df; nr += m*m;
    }
    const float c = gcnt[g];
    sAdd(&s_msq[t], c*nm);
    sAdd(&s_rB[t],  c*nr);
  }
#endif
  __syncthreads();

  // ||test_mean||^2
  const float tc = gtsc[0];
  float part = 0.f;
  for (int d = tid; d < D_; d += blockDim.x) {
    const float m = gtsum[d] / fmaxf(1.f, tc);
    part += m*m;
  }
  s_red[tid] = part;
  __syncthreads();
  for (int s = 128; s > 0; s >>= 1) {
    if (tid < s) s_red[tid] += s_red[tid + s];
    __syncthreads();
  }
  if (tid < T_) {
    const float test_sqn = gtsc[1] - tc * s_red[0];          // sum ||x - tm||^2 over test
    const float test_var = test_sqn / fmaxf(1.f, tc - 1.f);
    const float dn  = fmaxf(1.f, s_tcnt[tid] - 1.f);
    const float msq = s_msq[tid] / dn;
    const float rsq = (s_rA[tid] - s_rB[tid]) / dn;          // sum||x-mu||^2 = sum||x||^2 - cnt||mu||^2
    const float asq = (test_var - msq) / fmaxf(1e-6f, rsq);
    galpha[tid] = (asq > 0.f) ? sqrtf(fmaxf(asq, 0.f)) : 0.f;
  }
}

// ---------------------------------------------------------------------------
// Pass 3: out = train ? a*x + (1-a)*mean[seg] : x  (1 GB streaming, NT hints)
// ---------------------------------------------------------------------------
__global__ __launch_bounds__(256) void jj_norm_p3(
    const float* __restrict__ x, const int* __restrict__ labels,
    const int* __restrict__ times, const float* __restrict__ gmean,
    const float* __restrict__ galpha, float* __restrict__ out, int N)
{
  __shared__ float s_alpha[T_];
  if (threadIdx.x < T_) s_alpha[threadIdx.x] = galpha[threadIdx.x];
  __syncthreads();
  const int lane = threadIdx.x & 31;
  const int wv   = threadIdx.x >> 5;
  const int stride = gridDim.x * 8;
  for (int row = blockIdx.x * 8 + wv; row < N; row += stride) {
    const size_t base = (size_t)row * D_ + lane * 4;      // wave32 x float4 = full row
    v4f xv = __builtin_nontemporal_load((const v4f*)(x + base));
    const int t = times[row];                              // wave-uniform branch
    v4f o = xv;
    if (t < SPLIT_) {
      const int g = t * L_ + labels[row];
      const v4f m = *(const v4f*)(gmean + (size_t)g * D_ + lane * 4);  // L2-resident table
      const float a = s_alpha[t], b = 1.f - a;
      o[0] = a*xv[0] + b*m[0];
      o[1] = a*xv[1] + b*m[1];
      o[2] = a*xv[2] + b*m[2];
      o[3] = a*xv[3] + b*m[3];
    }
    __builtin_nontemporal_store(o, (v4f*)(out + base));
    if (row + stride < N)
      __builtin_prefetch(x + (size_t)(row + stride) * D_ + lane * 4, 0, 0);
  }
}

extern "C" void kernel_launch(void* const* d_in, const int* in_sizes, int n_in,
                              void* d_out, int out_size, void* d_ws, size_t ws_size,
                              hipStream_t stream) {
  const float* x      = (const float*)d_in[0];
  const int*   labels = (const int*)d_in[1];
  const int*   times  = (const int*)d_in[2];
  const int N = in_sizes[1];           // 1,000,000 rows

  // Workspace layout (floats): needs ~827 KB
  float* ws     = (float*)d_ws;
  float* gsums  = ws;                  // 800*128
  float* gsumsq = ws + 102400;         // 800
  float* gcnt   = ws + 103200;         // 800
  float* gtsum  = ws + 104000;         // 128
  float* gtsc   = ws + 104128;         // 2
  float* gmean  = ws + 104192;         // 800*128
  float* galpha = ws + 206592;         // 20

  // Zero accumulators every call (deterministic, graph-capturable)
  hipMemsetAsync(ws, 0, (size_t)104130 * sizeof(float), stream);

  const size_t lds1 = (size_t)SM_TOTAL * sizeof(float);   // ~117 KB (2 blocks/WGP)
  dim3 g1(128, NSLICE);
  jj_norm_p1<<<g1, 256, lds1, stream>>>(x, labels, times, N, gsums, gsumsq, gcnt, gtsum, gtsc);
  jj_norm_p2<<<1, 256, 0, stream>>>(gsums, gsumsq, gcnt, gtsum, gtsc, gmean, galpha);
  jj_norm_p3<<<2048, 256, 0, stream>>>(x, labels, times, gmean, galpha, (float*)d_out, N);
}